// LMC_27736898798357
// MI455X (gfx1250) — compile-verified
//
#include <hip/hip_runtime.h>
#include <cstdint>

// ---------------- problem constants (match reference) ----------------
constexpr int   NUM_RAYS  = 65536;
constexpr int   NUM_IMGS  = 64;
constexpr int   N_PER_IMG = NUM_RAYS / NUM_IMGS;   // 1024
constexpr int   Hc        = 800;
constexpr int   Wc        = 800;
constexpr int   ROW       = Hc * Wc + 1;           // 640001 cdf entries per image
constexpr int   REINIT    = 6553;                  // int(0.1*65536)
constexpr int   U_NUM     = 6553;
constexpr int   KSEL      = REINIT + 1;            // k-th smallest (1-indexed)
constexpr int   COARSE    = 1024;                  // coarse segments
constexpr int   SEG       = (ROW - 1) / COARSE;    // 625
constexpr float Af        = 20.0f;
constexpr float Bf        = 0.02f;

// ------------------------------------------------------------------
// Kernel 1: exact k-th smallest of loss via 4-pass radix select.
// Single block, 1024 threads. float4 loads, LDS histogram (ds_add),
// 256-wide Hillis-Steele scan to pick the bucket (no serial loop).
// ------------------------------------------------------------------
__global__ __launch_bounds__(1024) void LMC_select_thr(const float* __restrict__ loss,
                                                       float* __restrict__ ws) {
    __shared__ unsigned int hist[256];
    __shared__ unsigned int scan[256];
    __shared__ unsigned int sh_prefix;
    __shared__ unsigned int sh_k;
    const int tid = threadIdx.x;
    if (tid == 0) { sh_prefix = 0u; sh_k = (unsigned int)KSEL; }
    __syncthreads();

    const float4* __restrict__ l4 = (const float4*)loss;

    for (int shift = 24; shift >= 0; shift -= 8) {
        const unsigned int prefix = sh_prefix;       // captured after barrier
        const unsigned int kcur   = sh_k;
        if (tid < 256) hist[tid] = 0u;
        __syncthreads();

        for (int i = tid; i < NUM_RAYS / 4; i += 1024) {
            float4 v = l4[i];
            const float vv[4] = {v.x, v.y, v.z, v.w};
            #pragma unroll
            for (int c = 0; c < 4; ++c) {
                unsigned int bits = __float_as_uint(vv[c]);
                unsigned int key  = (bits & 0x80000000u) ? ~bits : (bits | 0x80000000u);
                bool match = (shift == 24) ||
                             ((key >> (shift + 8)) == (prefix >> (shift + 8)));
                if (match) atomicAdd(&hist[(key >> shift) & 255u], 1u);
            }
        }
        __syncthreads();

        // parallel inclusive scan of the 256-bin histogram
        if (tid < 256) scan[tid] = hist[tid];
        __syncthreads();
        #pragma unroll
        for (int d = 1; d < 256; d <<= 1) {
            unsigned int v = 0u;
            if (tid < 256) { v = scan[tid]; if (tid >= d) v += scan[tid - d]; }
            __syncthreads();
            if (tid < 256) scan[tid] = v;
            __syncthreads();
        }
        if (tid < 256) {
            unsigned int incl = scan[tid];
            unsigned int excl = incl - hist[tid];
            if (excl < kcur && kcur <= incl) {       // exactly one thread matches
                sh_prefix = prefix | ((unsigned int)tid << shift);
                sh_k      = kcur - excl;
            }
        }
        __syncthreads();
    }
    if (tid == 0) {
        unsigned int key  = sh_prefix;
        unsigned int bits = (key & 0x80000000u) ? (key ^ 0x80000000u) : ~key;
        ws[0] = __uint_as_float(bits);
    }
}

// ------------------------------------------------------------------
// Kernel 2: one block per image (1024 threads = 32 waves).
//  - CDNA5 async global->LDS gather of the coarse CDF level,
//    overlapped with per-ray ALU work; s_wait_asynccnt + barrier
//  - level 1: binary search in LDS (1025 entries, 10 cheap LDS rounds)
//  - level 2: QUATERNARY search in the 625-entry window: 3 independent
//    probes per round -> ~5 dependent memory rounds, + counting tail
//  - wave32 ballot+popc prefix sum of `bound`, LDS wave-total scan
//  - float2 vector loads/stores for all per-ray data
// ------------------------------------------------------------------
__global__ __launch_bounds__(1024) void LMC_resample(const float2* __restrict__ net_grad,
                                                     const float* __restrict__ loss,
                                                     const float2* __restrict__ prev,
                                                     const float* __restrict__ cdf,
                                                     const float2* __restrict__ noise,
                                                     const float2* __restrict__ rand_ten,
                                                     const float* __restrict__ u,
                                                     const float* __restrict__ ws,
                                                     float2* __restrict__ out) {
    __shared__ float        coarse[COARSE + 1];
    __shared__ int          sidx[N_PER_IMG];
    __shared__ unsigned int wtot[32];

    const int tid = threadIdx.x;
    const int img = blockIdx.x;
    const int r   = img * N_PER_IMG + tid;
    const float* __restrict__ crow = cdf + (size_t)img * ROW;

    // ---- CDNA5 async global->LDS gather of the coarse CDF level ----
    {
        unsigned int lds_a = (unsigned int)(uintptr_t)(&coarse[tid]);
        const float* g     = crow + (size_t)tid * SEG;
        asm volatile("global_load_async_to_lds_b32 %0, %1, off"
                     :: "v"(lds_a), "v"(g) : "memory");
        if (tid == 0) {
            unsigned int lds_b = (unsigned int)(uintptr_t)(&coarse[COARSE]);
            const float* g2    = crow + (size_t)COARSE * SEG;   // entry 640000
            asm volatile("global_load_async_to_lds_b32 %0, %1, off"
                         :: "v"(lds_b), "v"(g2) : "memory");
        }
    }

    // ---- overlap: per-ray ALU work while the async DMA is in flight ----
    const float  thr = ws[0];
    const float2 g2v = net_grad[r];
    const float2 nzv = noise[r];
    const float2 pvv = prev[r];
    const float  lp  = loss[r];
    const float  q   = u[r];                     // u is (64,1024) flat == r

    const float ps0 = pvv.x + g2v.x * Af + nzv.x * Bf;
    const float ps1 = pvv.y + g2v.y * Af + nzv.y * Bf;
    const bool bound = (ps0 < 0.0f) | (ps1 < 0.0f) |
                       (ps0 > 1.0f) | (ps1 > 1.0f) | (lp <= thr);

    asm volatile("s_wait_asynccnt 0" ::: "memory");
    __syncthreads();

    // ---- level 1: binary search of q in LDS coarse[0..1024] (right side) ----
    int lo = 0, hi = COARSE + 1;                 // insertion point in [0,1025]
    while (lo < hi) {
        int mid = (lo + hi) >> 1;
        if (coarse[mid] <= q) lo = mid + 1; else hi = mid;
    }
    int jc = lo;
    if (jc < 1) jc = 1;                          // q >= 0 == coarse[0] anyway

    // ---- level 2: quaternary search, 3 independent probes per round ----
    int flo = (jc - 1) * SEG;
    int fhi = (jc <= COARSE) ? jc * SEG : ROW;   // window size <= 625
    while (fhi - flo > 4) {
        const int s  = (fhi - flo) >> 2;
        const int m1 = flo + s, m2 = flo + 2 * s, m3 = flo + 3 * s;
        const float c1 = crow[m1];               // three independent loads:
        const float c2 = crow[m2];               // one memory round per iter
        const float c3 = crow[m3];
        if (c1 > q)        { fhi = m1; }
        else if (c2 > q)   { flo = m1 + 1; fhi = m2; }
        else if (c3 > q)   { flo = m2 + 1; fhi = m3; }
        else               { flo = m3 + 1; }
    }
    // counting tail over <=4 sorted entries (independent loads)
    int cnt = (crow[flo] <= q) ? 1 : 0;
    if (flo + 1 < fhi) cnt += (crow[flo + 1] <= q) ? 1 : 0;
    if (flo + 2 < fhi) cnt += (crow[flo + 2] <= q) ? 1 : 0;
    if (flo + 3 < fhi) cnt += (crow[flo + 3] <= q) ? 1 : 0;
    int idx = (flo + cnt) - 1;                   // searchsorted(...,'right') - 1
    idx = min(max(idx, 0), Hc * Wc - 1);
    sidx[tid] = idx;

    // ---- wave32 ballot prefix-sum of bound across the 1024-ray image ----
    unsigned long long mask = __ballot(bound);   // wave32: low 32 bits valid
    const int lane = tid & 31, wid = tid >> 5;
    unsigned int incl = (unsigned int)__popcll(mask & (((unsigned long long)2u << lane) - 1ull));
    if (lane == 0) wtot[wid] = (unsigned int)__popcll(mask);
    __syncthreads();                             // covers sidx[] and wtot[]

    unsigned int off = 0;
    for (int w = 0; w < wid; ++w) off += wtot[w];
    int pos = (int)(off + incl) - 1;             // cumsum - 1
    pos = min(max(pos, 0), N_PER_IMG - 1);

    const int vals = sidx[pos];
    const float ny = (float)(vals / Wc) / (float)(Hc - 1);
    const float nx = (float)(vals % Wc) / (float)(Wc - 1);

    // last U_NUM rays were overwritten with rand_ten before the where()
    float b0 = ps0, b1 = ps1;
    if (r >= NUM_RAYS - U_NUM) { const float2 rt = rand_ten[r]; b0 = rt.x; b1 = rt.y; }

    float o0 = bound ? ny : b0;
    float o1 = bound ? nx : b1;
    o0 = fminf(fmaxf(o0, 0.0f), 1.0f);
    o1 = fminf(fmaxf(o1, 0.0f), 1.0f);

    // jnp.round is half-to-even; rintf under default RNE mode matches
    float2 o;
    o.x = rintf(o0 * (float)(Hc - 1));
    o.y = rintf(o1 * (float)(Wc - 1));
    out[r] = o;
}

extern "C" void kernel_launch(void* const* d_in, const int* in_sizes, int n_in,
                              void* d_out, int out_size, void* d_ws, size_t ws_size,
                              hipStream_t stream) {
    const float2* net_grad = (const float2*)d_in[0];
    const float*  loss     = (const float*)d_in[1];
    const float2* prev     = (const float2*)d_in[2];
    const float*  cdf      = (const float*)d_in[3];
    const float2* noise    = (const float2*)d_in[4];
    const float2* rand_ten = (const float2*)d_in[5];
    const float*  u        = (const float*)d_in[6];
    float* ws   = (float*)d_ws;
    float2* out = (float2*)d_out;

    LMC_select_thr<<<1, 1024, 0, stream>>>(loss, ws);
    LMC_resample<<<NUM_IMGS, N_PER_IMG, 0, stream>>>(net_grad, loss, prev, cdf,
                                                     noise, rand_ten, u, ws, out);
}